// StructureRefiner_86870008529458
// MI455X (gfx1250) — compile-verified
//
#include <hip/hip_runtime.h>
#include <math.h>

// ---------------- problem constants ----------------
#define BATCH 16
#define SEQL  1024
#define FDIM  320
#define HDIM  160
#define NJ    4            // j-chunks for the pairwise-grad kernel
#define JCH   (SEQL / NJ)  // 256
#define NSTEPS 100

#define LRATE 0.01f
#define BETA1 0.9f
#define BETA2 0.999f
#define ADAM_EPS 1e-8f

typedef float v2f __attribute__((ext_vector_type(2)));
typedef float v8f __attribute__((ext_vector_type(8)));

// ---------------- GEMM1: X[16384,320] @ W1b[320,160] + b, relu  (WMMA f32 16x16x4)
// One wave computes one 16x16 output tile. K loop = 320/4 = 80 WMMAs.
__global__ void __launch_bounds__(256)
gemm1_wmma(const float* __restrict__ X, const float* __restrict__ W,
           const float* __restrict__ bias, float* __restrict__ Hout) {
  const int M = BATCH * SEQL;            // 16384
  const int K = FDIM, N = HDIM;          // 320, 160
  const int NT = N / 16;                 // 10 n-tiles
  int wave = (blockIdx.x * blockDim.x + threadIdx.x) >> 5;
  int lane = threadIdx.x & 31;
  int mt = wave / NT;
  int nt = wave - mt * NT;
  if (mt >= M / 16) return;              // wave-uniform guard
  int m0 = mt * 16, n0 = nt * 16;
  int lr = lane & 15;                    // 0..15
  int hi = lane >> 4;                    // 0/1 half-select

  v8f c = {};
  const float* xrow = X + (size_t)(m0 + lr) * K;
  for (int k = 0; k < K; k += 4) {
    // A 16x4 (MxK): lanes0-15 hold K=0,1 ; lanes16-31 hold K=2,3 (ISA §7.12.2)
    v2f a;
    a.x = xrow[k + 2 * hi + 0];
    a.y = xrow[k + 2 * hi + 1];
    // B 4x16 (KxN): row striped across lanes within a VGPR; halves split K
    v2f b;
    b.x = W[(size_t)(k + 2 * hi + 0) * N + n0 + lr];
    b.y = W[(size_t)(k + 2 * hi + 1) * N + n0 + lr];
    c = __builtin_amdgcn_wmma_f32_16x16x4_f32(false, a, false, b,
                                              (short)0, c, false, false);
  }
  // C/D layout: VGPR r -> M = r (lanes 0-15) / M = 8+r (lanes 16-31); N = lane%16
  float bv = bias[n0 + lr];
#pragma unroll
  for (int r = 0; r < 8; ++r) {
    float val = c[r] + bv;
    val = val > 0.f ? val : 0.f;         // relu
    Hout[(size_t)(m0 + r + 8 * hi) * N + n0 + lr] = val;
  }
}

// ---------------- GEMM2: H[16384,160] @ W2b[160,3] + b2  -> structure [16384,3]
__global__ void __launch_bounds__(256)
gemm2_small(const float* __restrict__ Hbuf, const float* __restrict__ W2,
            const float* __restrict__ b2, float* __restrict__ S) {
  int row = blockIdx.x * blockDim.x + threadIdx.x;
  if (row >= BATCH * SEQL) return;
  const float* h = Hbuf + (size_t)row * HDIM;
  float a0 = b2[0], a1 = b2[1], a2 = b2[2];
#pragma unroll 4
  for (int k = 0; k < HDIM; ++k) {
    float hv = h[k];
    a0 = fmaf(hv, W2[k * 3 + 0], a0);
    a1 = fmaf(hv, W2[k * 3 + 1], a1);
    a2 = fmaf(hv, W2[k * 3 + 2], a2);
  }
  S[row * 3 + 0] = a0; S[row * 3 + 1] = a1; S[row * 3 + 2] = a2;
}

// ---------------- one-time: csym = C + C^T  (LDS-tiled, both streams coalesced)
__global__ void __launch_bounds__(256)
symmetrize_cm(const float* __restrict__ CM, float* __restrict__ CS) {
  // grid: (L/32, L/32, B); block (32, 8)
  __shared__ float tb[32][33];           // padded: conflict-free transposed read
  int b  = blockIdx.z;
  int i0 = blockIdx.x * 32;              // output row tile
  int j0 = blockIdx.y * 32;              // output col tile
  int tx = threadIdx.x, ty = threadIdx.y;
  const float* base = CM + (size_t)b * SEQL * SEQL;
  for (int r = ty; r < 32; r += 8)       // tile of C^T source: rows j0.., cols i0..
    tb[r][tx] = base[(size_t)(j0 + r) * SEQL + i0 + tx];
  __syncthreads();
  float* outb = CS + (size_t)b * SEQL * SEQL;
  for (int r = ty; r < 32; r += 8) {
    float a = base[(size_t)(i0 + r) * SEQL + j0 + tx];      // C[b, i, j]
    outb[(size_t)(i0 + r) * SEQL + j0 + tx] = a + tb[tx][r]; // + C[b, j, i]
  }
}

// ---------------- per-step contact-loss gradient (partial over a j-chunk)
// g_i = (1/(B*L*L)) * sum_j w * (s_i - s_j) * rsqrt(sq)
// w = csum*([sq>64] + [sq<144]) - 2*[sq<144], csum = c_ij + c_ji (precomputed)
// sq==0 -> zero grad (safe_norm); strict compares match jax relu'(0)=0.
__global__ void __launch_bounds__(256)
grad_contact(const float* __restrict__ S, const float* __restrict__ CS,
             float* __restrict__ gpart) {
  int b  = blockIdx.z;
  int it = blockIdx.x;
  int jc = blockIdx.y;
  int tid = threadIdx.x;
  int i  = it * 256 + tid;
  int j0 = jc * JCH;

  __shared__ float sj[JCH * 3];
  const float* sb = S + (size_t)b * SEQL * 3;
  for (int k = tid; k < JCH * 3; k += 256) sj[k] = sb[j0 * 3 + k];
  __syncthreads();

  float six = sb[i * 3 + 0], siy = sb[i * 3 + 1], siz = sb[i * 3 + 2];
  const float* crow = CS + ((size_t)b * SEQL + i) * SEQL + j0; // csym[b,i,j0..]

  float gx = 0.f, gy = 0.f, gz = 0.f;
  for (int jj = 0; jj < JCH; jj += 4) {
    float4 c4 = *reinterpret_cast<const float4*>(crow + jj);   // global_load_b128
    float cc[4] = {c4.x, c4.y, c4.z, c4.w};
#pragma unroll
    for (int u = 0; u < 4; ++u) {
      int jl = jj + u;
      float dx = six - sj[jl * 3 + 0];
      float dy = siy - sj[jl * 3 + 1];
      float dz = siz - sj[jl * 3 + 2];
      float sq = fmaf(dx, dx, fmaf(dy, dy, dz * dz));
      // d>8 <=> sq>64 ; d<12 <=> sq<144
      float fa = sq > 64.f  ? 1.f : 0.f;
      float fb = sq < 144.f ? 1.f : 0.f;
      float w  = fmaf(cc[u], fa + fb, -2.f * fb);
      float scale = (sq > 0.f) ? w * __builtin_amdgcn_rsqf(sq) : 0.f;
      gx = fmaf(scale, dx, gx);
      gy = fmaf(scale, dy, gy);
      gz = fmaf(scale, dz, gz);
    }
  }
  const float cnorm = 1.f / ((float)BATCH * (float)SEQL * (float)SEQL);
  float* gp = gpart + ((size_t)((b * NJ + jc) * SEQL) + i) * 3;
  gp[0] = gx * cnorm; gp[1] = gy * cnorm; gp[2] = gz * cnorm;
}

// ---------------- Adam update: reduce grad partials, add bond grad, step.
__global__ void __launch_bounds__(256)
adam_step(const float* __restrict__ Scur, float* __restrict__ Snext,
          float* __restrict__ m, float* __restrict__ v,
          const float* __restrict__ gpart, float c1, float c2) {
  int idx = blockIdx.x * blockDim.x + threadIdx.x;   // (b,i) pair
  if (idx >= BATCH * SEQL) return;
  int i = idx & (SEQL - 1);
  int b = idx >> 10;

  float g[3];
#pragma unroll
  for (int k = 0; k < 3; ++k) {
    float acc = 0.f;
#pragma unroll
    for (int jc = 0; jc < NJ; ++jc)
      acc += gpart[((size_t)((b * NJ + jc) * SEQL) + i) * 3 + k];
    g[k] = acc;
  }

  const float* srow = Scur + (size_t)idx * 3;
  float sx = srow[0], sy = srow[1], sz = srow[2];
  const float bnorm = 0.1f / ((float)BATCH * (float)(SEQL - 1));
  if (i > 0) {                       // bond (i-1, i):  +(s_i - s_{i-1})/d if d>4
    float dx = sx - srow[-3], dy = sy - srow[-2], dz = sz - srow[-1];
    float sq = dx * dx + dy * dy + dz * dz;
    if (sq > 16.f) {
      float r = __builtin_amdgcn_rsqf(sq) * bnorm;
      g[0] = fmaf(dx, r, g[0]); g[1] = fmaf(dy, r, g[1]); g[2] = fmaf(dz, r, g[2]);
    }
  }
  if (i < SEQL - 1) {                // bond (i, i+1):  +(s_i - s_{i+1})/d if d>4
    float dx = sx - srow[3], dy = sy - srow[4], dz = sz - srow[5];
    float sq = dx * dx + dy * dy + dz * dz;
    if (sq > 16.f) {
      float r = __builtin_amdgcn_rsqf(sq) * bnorm;
      g[0] = fmaf(dx, r, g[0]); g[1] = fmaf(dy, r, g[1]); g[2] = fmaf(dz, r, g[2]);
    }
  }

  float s3[3] = {sx, sy, sz};
#pragma unroll
  for (int k = 0; k < 3; ++k) {
    size_t o = (size_t)idx * 3 + k;
    float mm = BETA1 * m[o] + (1.f - BETA1) * g[k];
    float vv = BETA2 * v[o] + (1.f - BETA2) * g[k] * g[k];
    m[o] = mm; v[o] = vv;
    float upd = LRATE * (mm * c1) / (sqrtf(vv * c2) + ADAM_EPS);
    Snext[o] = s3[k] - upd;
  }
}

__global__ void zero_f32(float* __restrict__ p, int n) {
  int i = blockIdx.x * blockDim.x + threadIdx.x;
  if (i < n) p[i] = 0.f;
}

// ---------------- host orchestration ----------------
extern "C" void kernel_launch(void* const* d_in, const int* in_sizes, int n_in,
                              void* d_out, int out_size, void* d_ws, size_t ws_size,
                              hipStream_t stream) {
  const float* bb  = (const float*)d_in[0];   // backbone_features [16,1024,320]
  const float* cm  = (const float*)d_in[2];   // contact_map [16,1024,1024]
  const float* W1b = (const float*)d_in[3];
  const float* b1b = (const float*)d_in[4];
  const float* W2b = (const float*)d_in[5];
  const float* b2b = (const float*)d_in[6];
  float* out = (float*)d_out;                 // [16,1024,3]

  const int NS = BATCH * SEQL * 3;            // 49152
  float* s0    = (float*)d_ws;                // structure buf A
  float* s1    = s0 + NS;                     // structure buf B
  float* mbuf  = s1 + NS;                     // Adam m
  float* vbuf  = mbuf + NS;                   // Adam v
  float* gpart = vbuf + NS;                   // [16, NJ, 1024, 3]
  float* Hbuf  = gpart + (size_t)BATCH * NJ * SEQL * 3;      // [16384,160]
  float* csym  = Hbuf + (size_t)BATCH * SEQL * HDIM;         // [16,1024,1024] 64MB

  // zero Adam state (m and v are contiguous)
  zero_f32<<<(2 * NS + 255) / 256, 256, 0, stream>>>(mbuf, 2 * NS);

  // one-time: symmetrized contact map (stays L2-resident: 64MB < 192MB L2)
  symmetrize_cm<<<dim3(SEQL / 32, SEQL / 32, BATCH), dim3(32, 8), 0, stream>>>(cm, csym);

  // backbone MLP: WMMA GEMM1 then skinny GEMM2
  {
    int waves  = (BATCH * SEQL / 16) * (HDIM / 16);   // 1024 * 10
    int blocks = waves * 32 / 256;                    // 1280
    gemm1_wmma<<<blocks, 256, 0, stream>>>(bb, W1b, b1b, Hbuf);
    gemm2_small<<<(BATCH * SEQL + 255) / 256, 256, 0, stream>>>(Hbuf, W2b, b2b, s0);
  }

  // 100 Adam steps, double-buffered structure
  float* cur = s0;
  float* nxt = s1;
  for (int t = 1; t <= NSTEPS; ++t) {
    grad_contact<<<dim3(SEQL / 256, NJ, BATCH), 256, 0, stream>>>(cur, csym, gpart);
    float c1 = 1.0f / (1.0f - powf(BETA1, (float)t));
    float c2 = 1.0f / (1.0f - powf(BETA2, (float)t));
    float* dst = (t == NSTEPS) ? out : nxt;
    adam_step<<<(BATCH * SEQL + 255) / 256, 256, 0, stream>>>(cur, dst, mbuf, vbuf,
                                                              gpart, c1, c2);
    float* tmp = cur; cur = nxt; nxt = tmp;
  }
}